// GraphGeneratorX_3341484556437
// MI455X (gfx1250) — compile-verified
//
#include <hip/hip_runtime.h>
#include <hip/hip_bf16.h>
#include <math.h>

typedef float v2f __attribute__((ext_vector_type(2)));
typedef float v8f __attribute__((ext_vector_type(8)));

// ---- helpers -------------------------------------------------------------
__device__ __forceinline__ unsigned int encOrd(float f) {
  unsigned int u = __float_as_uint(f);
  return (u & 0x80000000u) ? ~u : (u | 0x80000000u);   // monotone float->uint
}
__device__ __forceinline__ float decOrd(unsigned int k) {
  unsigned int u = (k & 0x80000000u) ? (k ^ 0x80000000u) : ~k;
  return __uint_as_float(u);
}
__device__ __forceinline__ float eluf(float x)    { return x > 0.f ? x : expm1f(x); }
__device__ __forceinline__ float sigm(float x)    { return 1.f / (1.f + expf(-x)); }

// ---- zero ----------------------------------------------------------------
__global__ void k_zero(float* __restrict__ p, int n) {
  for (int i = blockIdx.x * blockDim.x + threadIdx.x; i < n; i += gridDim.x * blockDim.x)
    p[i] = 0.f;
}

// ---- edge scatter: 2 channels -------------------------------------------
__global__ void k_scatter2(const float* __restrict__ x, const int* __restrict__ src,
                           const int* __restrict__ dst, float* __restrict__ agg, int E) {
  int e = blockIdx.x * blockDim.x + threadIdx.x;
  if (e >= E) return;
  if (e + 8192 < E) {                     // stream-ahead prefetch of index arrays
    __builtin_prefetch(src + e + 8192, 0, 0);
    __builtin_prefetch(dst + e + 8192, 0, 0);
  }
  int s = src[e], d = dst[e];
  float2 v = ((const float2*)x)[s];
  atomicAdd(&agg[(size_t)d * 2 + 0], v.x);
  atomicAdd(&agg[(size_t)d * 2 + 1], v.y);
}

// ---- edge scatter: 8 channels -------------------------------------------
__global__ void k_scatter8(const float* __restrict__ h, const int* __restrict__ src,
                           const int* __restrict__ dst, float* __restrict__ agg, int E) {
  int e = blockIdx.x * blockDim.x + threadIdx.x;
  if (e >= E) return;
  if (e + 8192 < E) {
    __builtin_prefetch(src + e + 8192, 0, 0);
    __builtin_prefetch(dst + e + 8192, 0, 0);
  }
  int s = src[e], d = dst[e];
  const float4* hv = (const float4*)(h + (size_t)s * 8);
  float4 a = hv[0], b = hv[1];
  float* base = agg + (size_t)d * 8;
  atomicAdd(base + 0, a.x); atomicAdd(base + 1, a.y);
  atomicAdd(base + 2, a.z); atomicAdd(base + 3, a.w);
  atomicAdd(base + 4, b.x); atomicAdd(base + 5, b.y);
  atomicAdd(base + 6, b.z); atomicAdd(base + 7, b.w);
}

// ---- layer1: [N,2] -> elu -> [N,8] --------------------------------------
__global__ void k_layer1(const float* __restrict__ x, const float* __restrict__ agg,
                         const float* __restrict__ wr, const float* __restrict__ wsl,
                         const float* __restrict__ b, float* __restrict__ h, int n) {
  int i = blockIdx.x * blockDim.x + threadIdx.x;
  if (i >= n) return;
  float2 xv = ((const float2*)x)[i];
  float2 av = ((const float2*)agg)[i];
  float r[8];
#pragma unroll
  for (int k = 0; k < 8; ++k) {
    float s = b[k] + av.x * wr[k * 2] + av.y * wr[k * 2 + 1]
                   + xv.x * wsl[k * 2] + xv.y * wsl[k * 2 + 1];
    r[k] = eluf(s);
  }
  float4* hv = (float4*)(h + (size_t)i * 8);
  hv[0] = make_float4(r[0], r[1], r[2], r[3]);
  hv[1] = make_float4(r[4], r[5], r[6], r[7]);
}

// ---- layer2 + global max pool, via V_WMMA_F32_16X16X4_F32 ----------------
// out[m,n] = sum_k agg8[m,k]*w2r[n,k] + sum_k h1[m,k]*w2s[n,k] + b2[n]
__global__ void k_layer2_pool_wmma(const float* __restrict__ h1, const float* __restrict__ agg8,
                                   const float* __restrict__ w2r, const float* __restrict__ w2s,
                                   const float* __restrict__ b2, unsigned int* __restrict__ featKeys,
                                   int numTiles) {
  const int lane = threadIdx.x & 31;
  const int m    = lane & 15;
  const int kh   = lane >> 4;
  const int wave = blockIdx.x * (blockDim.x >> 5) + (threadIdx.x >> 5);
  const int nw   = gridDim.x * (blockDim.x >> 5);

  v2f bR0 = *(const v2f*)(w2r + m * 8 + 2 * kh);        // K 0..3
  v2f bR1 = *(const v2f*)(w2r + m * 8 + 4 + 2 * kh);    // K 4..7
  v2f bS0 = *(const v2f*)(w2s + m * 8 + 2 * kh);
  v2f bS1 = *(const v2f*)(w2s + m * 8 + 4 + 2 * kh);
  const float bias = b2[m];

  for (int t = wave; t < numTiles; t += nw) {
    const float* rA = agg8 + (size_t)(t * 16 + m) * 8;
    const float* rH = h1   + (size_t)(t * 16 + m) * 8;
    v2f aA0 = *(const v2f*)(rA + 2 * kh);
    v2f aA1 = *(const v2f*)(rA + 4 + 2 * kh);
    v2f aH0 = *(const v2f*)(rH + 2 * kh);
    v2f aH1 = *(const v2f*)(rH + 4 + 2 * kh);
    v8f acc = {};
    acc = __builtin_amdgcn_wmma_f32_16x16x4_f32(false, aA0, false, bR0, (short)0, acc, false, false);
    acc = __builtin_amdgcn_wmma_f32_16x16x4_f32(false, aA1, false, bR1, (short)0, acc, false, false);
    acc = __builtin_amdgcn_wmma_f32_16x16x4_f32(false, aH0, false, bS0, (short)0, acc, false, false);
    acc = __builtin_amdgcn_wmma_f32_16x16x4_f32(false, aH1, false, bS1, (short)0, acc, false, false);
    float mx = acc[0];
#pragma unroll
    for (int i = 1; i < 8; ++i) mx = fmaxf(mx, acc[i]);
    mx += bias;                             // bias uniform per channel -> add after max
    mx = fmaxf(mx, __shfl_xor(mx, 16));     // combine rows 0-7 with 8-15
    if (lane < 16) atomicMax(&featKeys[m], encOrd(mx));
  }
}

// scalar tail for N % 16 nodes (not hit for N=100000, kept for generality)
__global__ void k_layer2_pool_tail(const float* __restrict__ h1, const float* __restrict__ agg8,
                                   const float* __restrict__ w2r, const float* __restrict__ w2s,
                                   const float* __restrict__ b2, unsigned int* __restrict__ featKeys,
                                   int start, int n) {
  int i = start + blockIdx.x * blockDim.x + threadIdx.x;
  if (i >= n) return;
  const float* a = agg8 + (size_t)i * 8;
  const float* h = h1 + (size_t)i * 8;
  for (int k = 0; k < 16; ++k) {
    float s = b2[k];
    for (int j = 0; j < 8; ++j) s += a[j] * w2r[k * 8 + j] + h[j] * w2s[k * 8 + j];
    atomicMax(&featKeys[k], encOrd(s));
  }
}

// ---- layer3 via one V_WMMA_F32_16X16X4_F32 (K-concat trick) --------------
// out[m,n] = sum_{j<2} agg2[m,j]*w3r[n,j] + sum_{j<2} c[m,j]*w3s[n,j] + b3[n]
// A[m,0:2]=agg2[m,:], A[m,2:4]=c[m,:]  (lanes 0-15 load agg2, lanes 16-31 load c)
// B[0:2,n]=w3r[n,:],  B[2:4,n]=w3s[n,:]
// D: vgpr v holds row m = v + 8*(lane>>4), col n = lane&15 -> elu+bias, store h3[N,16]
__global__ void k_layer3_wmma(const float* __restrict__ cin, const float* __restrict__ agg2,
                              const float* __restrict__ w3r, const float* __restrict__ w3s,
                              const float* __restrict__ b3, float* __restrict__ h3,
                              int numTiles) {
  const int lane = threadIdx.x & 31;
  const int nL   = lane & 15;
  const int kh   = lane >> 4;
  const int wave = blockIdx.x * (blockDim.x >> 5) + (threadIdx.x >> 5);
  const int nw   = gridDim.x * (blockDim.x >> 5);

  const float* bsrc = kh ? w3s : w3r;
  v2f bF = *(const v2f*)(bsrc + nL * 2);
  const float bias = b3[nL];
  const float* asrc = kh ? cin : agg2;

  for (int t = wave; t < numTiles; t += nw) {
    v2f aF = *(const v2f*)(asrc + (size_t)(t * 16 + nL) * 2);
    v8f acc = {};
    acc = __builtin_amdgcn_wmma_f32_16x16x4_f32(false, aF, false, bF, (short)0, acc, false, false);
#pragma unroll
    for (int v = 0; v < 8; ++v) {
      int row = t * 16 + v + 8 * kh;
      h3[(size_t)row * 16 + nL] = eluf(acc[v] + bias);
    }
  }
}

// scalar tail for N % 16 nodes
__global__ void k_layer3_tail(const float* __restrict__ cin, const float* __restrict__ agg2,
                              const float* __restrict__ w3r, const float* __restrict__ w3s,
                              const float* __restrict__ b3, float* __restrict__ h3,
                              int start, int n) {
  int i = start + blockIdx.x * blockDim.x + threadIdx.x;
  if (i >= n) return;
  float2 xv = ((const float2*)cin)[i];
  float2 av = ((const float2*)agg2)[i];
  for (int k = 0; k < 16; ++k) {
    float s = b3[k] + av.x * w3r[k * 2] + av.y * w3r[k * 2 + 1]
                    + xv.x * w3s[k * 2] + xv.y * w3s[k * 2 + 1];
    h3[(size_t)i * 16 + k] = eluf(s);
  }
}

// ---- y' = h3 @ w4r[:, :16].T + prior @ w4r[:, 16:].T  (2 channels) -------
__global__ void k_yprime(const float* __restrict__ h3, const unsigned int* __restrict__ feat,
                         const float* __restrict__ w4r, float* __restrict__ y, int n) {
  int i = blockIdx.x * blockDim.x + threadIdx.x;
  if (i >= n) return;
  float p0 = 0.f, p1 = 0.f;
#pragma unroll
  for (int j = 0; j < 32; ++j) {
    float pv = decOrd(feat[j]);       // [featO(16), featM(16)] = prior
    p0 += pv * w4r[16 + j];
    p1 += pv * w4r[48 + 16 + j];
  }
  const float* hh = h3 + (size_t)i * 16;
  float y0 = p0, y1 = p1;
#pragma unroll
  for (int j = 0; j < 16; ++j) { float hv = hh[j]; y0 += hv * w4r[j]; y1 += hv * w4r[48 + j]; }
  ((float2*)y)[i] = make_float2(y0, y1);
}

// ---- final: sigmoid(aggY + b4 + h3@w4s.T + prior@w4s.T) + c --------------
__global__ void k_final(const float* __restrict__ h3, const float* __restrict__ aggY,
                        const unsigned int* __restrict__ feat, const float* __restrict__ w4s,
                        const float* __restrict__ b4, const float* __restrict__ cin,
                        float* __restrict__ out, int n) {
  int i = blockIdx.x * blockDim.x + threadIdx.x;
  if (i >= n) return;
  float q0 = 0.f, q1 = 0.f;
#pragma unroll
  for (int j = 0; j < 32; ++j) {
    float pv = decOrd(feat[j]);
    q0 += pv * w4s[16 + j];
    q1 += pv * w4s[48 + 16 + j];
  }
  const float* hh = h3 + (size_t)i * 16;
  float s0 = b4[0] + q0, s1 = b4[1] + q1;
#pragma unroll
  for (int j = 0; j < 16; ++j) { float hv = hh[j]; s0 += hv * w4s[j]; s1 += hv * w4s[48 + j]; }
  float2 av = ((const float2*)aggY)[i];
  s0 += av.x; s1 += av.y;
  float2 cv = ((const float2*)cin)[i];
  ((float2*)out)[i] = make_float2(sigm(s0) + cv.x, sigm(s1) + cv.y);
}

// ---- host ----------------------------------------------------------------
extern "C" void kernel_launch(void* const* d_in, const int* in_sizes, int n_in,
                              void* d_out, int out_size, void* d_ws, size_t ws_size,
                              hipStream_t stream) {
  const float* o  = (const float*)d_in[0];
  const float* m  = (const float*)d_in[1];
  const float* c  = (const float*)d_in[2];
  const int* eo   = (const int*)d_in[3];
  const int* em   = (const int*)d_in[4];
  const int* ec   = (const int*)d_in[5];
  const float* w1r = (const float*)d_in[6];
  const float* w1s = (const float*)d_in[7];
  const float* b1  = (const float*)d_in[8];
  const float* w2r = (const float*)d_in[9];
  const float* w2s = (const float*)d_in[10];
  const float* b2  = (const float*)d_in[11];
  const float* w3r = (const float*)d_in[12];
  const float* w3s = (const float*)d_in[13];
  const float* b3  = (const float*)d_in[14];
  const float* w4r = (const float*)d_in[15];
  const float* w4s = (const float*)d_in[16];
  const float* b4  = (const float*)d_in[17];

  const int N = in_sizes[0] / 2;
  const int E = in_sizes[3] / 2;

  float* W    = (float*)d_ws;                  // layout (floats):
  float* agg2 = W;                             // [0, 2N)
  float* agg8 = W + (size_t)2 * N;             // [2N, 10N)   (contiguous with agg2)
  float* h1   = W + (size_t)10 * N;            // [10N, 18N)
  float* h3   = W + (size_t)2 * N;             // aliases agg8+h1 (c-path only)
  float* yv   = W + (size_t)18 * N;            // [18N, 20N)
  float* aggY = W + (size_t)20 * N;            // [20N, 22N)
  unsigned int* feat = (unsigned int*)(W + (size_t)22 * N);  // 32 keys

  const int TPB = 256;
  const int EB  = (E + TPB - 1) / TPB;
  const int NB  = (N + TPB - 1) / TPB;
  const int ZB  = 2048;
  const int tiles = N / 16;
  const int rem   = N - tiles * 16;

  k_zero<<<ZB, TPB, 0, stream>>>((float*)feat, 32);   // keys: 0 == below any real float

  // ---- o path -> featO = feat[0:16] ----
  k_zero<<<ZB, TPB, 0, stream>>>(agg2, 10 * N);       // agg2 + agg8 in one pass
  k_scatter2<<<EB, TPB, 0, stream>>>(o, eo, eo + E, agg2, E);
  k_layer1<<<NB, TPB, 0, stream>>>(o, agg2, w1r, w1s, b1, h1, N);
  k_scatter8<<<EB, TPB, 0, stream>>>(h1, eo, eo + E, agg8, E);
  k_layer2_pool_wmma<<<160, TPB, 0, stream>>>(h1, agg8, w2r, w2s, b2, feat, tiles);
  if (rem > 0)
    k_layer2_pool_tail<<<1, TPB, 0, stream>>>(h1, agg8, w2r, w2s, b2, feat, tiles * 16, N);

  // ---- m path -> featM = feat[16:32] ----
  k_zero<<<ZB, TPB, 0, stream>>>(agg2, 10 * N);
  k_scatter2<<<EB, TPB, 0, stream>>>(m, em, em + E, agg2, E);
  k_layer1<<<NB, TPB, 0, stream>>>(m, agg2, w1r, w1s, b1, h1, N);
  k_scatter8<<<EB, TPB, 0, stream>>>(h1, em, em + E, agg8, E);
  k_layer2_pool_wmma<<<160, TPB, 0, stream>>>(h1, agg8, w2r, w2s, b2, feat + 16, tiles);
  if (rem > 0)
    k_layer2_pool_tail<<<1, TPB, 0, stream>>>(h1, agg8, w2r, w2s, b2, feat + 16, tiles * 16, N);

  // ---- c path ----
  k_zero<<<ZB, TPB, 0, stream>>>(agg2, 2 * N);
  k_scatter2<<<EB, TPB, 0, stream>>>(c, ec, ec + E, agg2, E);
  k_layer3_wmma<<<160, TPB, 0, stream>>>(c, agg2, w3r, w3s, b3, h3, tiles);
  if (rem > 0)
    k_layer3_tail<<<1, TPB, 0, stream>>>(c, agg2, w3r, w3s, b3, h3, tiles * 16, N);
  k_yprime<<<NB, TPB, 0, stream>>>(h3, feat, w4r, yv, N);
  k_zero<<<ZB, TPB, 0, stream>>>(aggY, 2 * N);
  k_scatter2<<<EB, TPB, 0, stream>>>(yv, ec, ec + E, aggY, E);   // 2-ch aggregation of projected feats
  k_final<<<NB, TPB, 0, stream>>>(h3, aggY, feat, w4s, b4, c, (float*)d_out, N);
}